// BPR_3582002725263
// MI455X (gfx1250) — compile-verified
//
#include <hip/hip_runtime.h>

// ----------------------------------------------------------------------------
// LightGCN 3-hop propagation on MI455X (gfx1250, wave32).
//   1) Build u-CSR and i-CSR on device (~2.5M u32 atomics total), edge
//      records interleaved as {src_index, weight} 8B pairs.
//   2) Run all 5 SpMMs as atomic-free "pull" passes (wave32 per dest row),
//      row gathers staged memory->LDS via the CDNA5 async engine
//      (global_load_async_to_lds_b64 + ASYNCcnt), depth-2 pipeline,
//      triple-buffered LDS ring.
//   3) Item-direction passes fuse the weighted-residual combine, so
//      gcn2_items / gcn3_items never touch memory.
// Working set ~140MB < 192MB L2 -> passes run at L2 bandwidth.
// ----------------------------------------------------------------------------

__device__ __forceinline__ void async_gather_b64(uint32_t lds_addr, const float* gptr) {
    // GLOBAL_LOAD_ASYNC_TO_LDS_B64 (GV mode): per-lane 8B memory->LDS,
    // tracked with ASYNCcnt, no VGPR writeback.
    asm volatile("global_load_async_to_lds_b64 %0, %1, off"
                 :: "v"(lds_addr), "v"(gptr)
                 : "memory");
}

template <int N>
__device__ __forceinline__ void wait_asynccnt() {
#if __has_builtin(__builtin_amdgcn_s_wait_asynccnt)
    __builtin_amdgcn_s_wait_asynccnt(N);
#else
    asm volatile("s_wait_asynccnt %0" :: "n"(N) : "memory");
#endif
}

// One wave32 per destination row; float2 per lane covers the 64-float row.
// Source rows are gathered memory->LDS with the async engine, depth-2
// pipeline over a 3-slot LDS ring (reuse distance 3 > prefetch depth 2).
// Edge records rec = {src, weight_bits} are loaded once at issue time;
// weights ride a 2-deep register pipeline to consumption.
__global__ void pull_kernel_async(int N, const int* __restrict__ off,
                                  const int2* __restrict__ recs,
                                  const float* __restrict__ X,
                                  float* __restrict__ Y,       // optional: store result rows
                                  float* __restrict__ outAcc,  // optional: out += alpha * result
                                  float alpha) {
    __shared__ __align__(16) float stage[8][3][64];   // 8 waves * 3 slots * 256B = 6KB
    const int wv   = (blockIdx.x * blockDim.x + threadIdx.x) >> 5;
    const int lane = threadIdx.x & 31;
    const int wib  = threadIdx.x >> 5;
    if (wv >= N) return;

    const int s = off[wv];
    const int e = off[wv + 1];
    const int col = lane * 2;

    float* ring = &stage[wib][0][0];
    // Low 32 bits of the generic (LDS-aperture) address = LDS byte offset.
    const uint32_t ring_lds = (uint32_t)(uintptr_t)ring + (uint32_t)col * 4u;

    float ax = 0.f, ay = 0.f;
    if (s < e) {
        // ---- prologue: fill pipeline to depth 2 ----
        int2 r0 = recs[s];
        async_gather_b64(ring_lds, X + (size_t)r0.x * 64 + col);
        float wcur = __int_as_float(r0.y);
        float wnxt = 0.f;
        int issued = 1;
        if (s + 1 < e) {
            int2 r1 = recs[s + 1];
            async_gather_b64(ring_lds + 256u, X + (size_t)r1.x * 64 + col);
            wnxt = __int_as_float(r1.y);
            issued = 2;
        }
        // ---- steady state ----
        for (int k = s; k < e; ++k) {
            if (e - k >= 2) wait_asynccnt<1>();   // load k done, k+1 in flight
            else            wait_asynccnt<0>();   // drain tail
            const int slot = (k - s) % 3;
            ax += wcur * ring[slot * 64 + col];
            ay += wcur * ring[slot * 64 + col + 1];
            float wnew = 0.f;
            if (s + issued < e) {
                int2 rn = recs[s + issued];
                async_gather_b64(ring_lds + (uint32_t)(issued % 3) * 256u,
                                 X + (size_t)rn.x * 64 + col);
                wnew = __int_as_float(rn.y);
                ++issued;
            }
            wcur = wnxt;
            wnxt = wnew;
        }
    }

    const size_t o = (size_t)wv * 64 + col;
    if (Y) {
        float2 r; r.x = ax; r.y = ay;
        *(float2*)(Y + o) = r;
    }
    if (outAcc) {
        float2* po = (float2*)(outAcc + o);
        float2 c = *po;
        c.x += alpha * ax;
        c.y += alpha * ay;
        *po = c;
    }
}

__global__ void zero_i32(int* __restrict__ a, int n) {
    int t = blockIdx.x * blockDim.x + threadIdx.x;
    if (t < n) a[t] = 0;
}

__global__ void hist_kernel(const int* __restrict__ u_idx, const int* __restrict__ i_idx,
                            int E, int* __restrict__ cnt_u, int* __restrict__ cnt_i) {
    int t = blockIdx.x * blockDim.x + threadIdx.x;
    if (t < E) {
        atomicAdd(&cnt_u[u_idx[t]], 1);
        atomicAdd(&cnt_i[i_idx[t]], 1);
    }
}

// Two independent exclusive scans (block 0: users, block 1: items).
// In-place: cnt[] is consumed chunk-by-chunk and rewritten as the cursor array.
__global__ void scan2_kernel(int* cnt_cur_u, int nu, int* off_u,
                             int* cnt_cur_i, int ni, int* off_i) {
    int* cnt = (blockIdx.x == 0) ? cnt_cur_u : cnt_cur_i;
    int* off = (blockIdx.x == 0) ? off_u : off_i;
    int  n   = (blockIdx.x == 0) ? nu : ni;
    __shared__ int buf[1024];
    const int tid = threadIdx.x;
    int base = 0;
    for (int start = 0; start < n; start += 1024) {
        int i = start + tid;
        int v = (i < n) ? cnt[i] : 0;
        buf[tid] = v;
        __syncthreads();
        for (int s = 1; s < 1024; s <<= 1) {
            int t = (tid >= s) ? buf[tid - s] : 0;
            __syncthreads();
            buf[tid] += t;
            __syncthreads();
        }
        int incl  = buf[tid];
        int total = buf[1023];
        __syncthreads();                 // before next chunk overwrites buf
        if (i < n) {
            int ex = base + incl - v;
            off[i] = ex;
            cnt[i] = ex;                 // becomes the scatter cursor
        }
        base += total;
    }
    if (tid == 0) off[n] = base;
}

__global__ void scatter_kernel(const int* __restrict__ u_idx, const int* __restrict__ i_idx,
                               const float* __restrict__ w, int E,
                               int* cur_u, int* cur_i,
                               int2* __restrict__ csr_u, int2* __restrict__ csr_i) {
    int t = blockIdx.x * blockDim.x + threadIdx.x;
    if (t < E) {
        int u = u_idx[t], i = i_idx[t];
        int wb = __float_as_int(w[t]);
        int p = atomicAdd(&cur_u[u], 1);
        int2 ru; ru.x = i; ru.y = wb;
        csr_u[p] = ru;
        int q = atomicAdd(&cur_i[i], 1);
        int2 ri; ri.x = u; ri.y = wb;
        csr_i[q] = ri;
    }
}

__global__ void copy_f4(const float4* __restrict__ src, float4* __restrict__ dst, int n4) {
    int t = blockIdx.x * blockDim.x + threadIdx.x;
    if (t < n4) dst[t] = src[t];
}

extern "C" void kernel_launch(void* const* d_in, const int* in_sizes, int n_in,
                              void* d_out, int out_size, void* d_ws, size_t ws_size,
                              hipStream_t stream) {
    (void)n_in; (void)out_size; (void)ws_size;
    const float* embed_user = (const float*)d_in[0];   // [U,64]
    const float* embed_item = (const float*)d_in[1];   // [I,64]
    const float* edge_vals  = (const float*)d_in[2];   // [E]
    const int*   u_idx      = (const int*)  d_in[3];   // [E]
    const int*   i_idx      = (const int*)  d_in[4];   // [E]

    const int D = 64;
    const int U = in_sizes[0] / D;
    const int I = in_sizes[1] / D;
    const int E = in_sizes[2];

    auto aln = [](size_t x) { return (x + (size_t)255) & ~(size_t)255; };
    char* p = (char*)d_ws;
    int*   cur_u = (int*)p;   p += aln((size_t)U * 4);
    int*   cur_i = (int*)p;   p += aln((size_t)I * 4);
    int*   off_u = (int*)p;   p += aln((size_t)(U + 1) * 4);
    int*   off_i = (int*)p;   p += aln((size_t)(I + 1) * 4);
    int2*  csr_u = (int2*)p;  p += aln((size_t)E * 8);
    int2*  csr_i = (int2*)p;  p += aln((size_t)E * 8);
    float* g1u   = (float*)p; p += aln((size_t)U * D * 4);
    float* g1i   = (float*)p; p += aln((size_t)I * D * 4);
    float* g2u   = (float*)p; p += aln((size_t)U * D * 4);
    float* out   = (float*)d_out;

    const int B = 256;                       // 8 wave32s per block
    auto cdiv = [](int a, int b) { return (a + b - 1) / b; };

    // ---- CSR build (both directions), rebuilt deterministically each call ----
    zero_i32<<<cdiv(U, B), B, 0, stream>>>(cur_u, U);
    zero_i32<<<cdiv(I, B), B, 0, stream>>>(cur_i, I);
    hist_kernel<<<cdiv(E, B), B, 0, stream>>>(u_idx, i_idx, E, cur_u, cur_i);
    scan2_kernel<<<2, 1024, 0, stream>>>(cur_u, U, off_u, cur_i, I, off_i);
    scatter_kernel<<<cdiv(E, B), B, 0, stream>>>(u_idx, i_idx, edge_vals, E,
                                                 cur_u, cur_i, csr_u, csr_i);

    // ---- out starts as the residual term (embed_item) ----
    copy_f4<<<cdiv(I * D / 4, B), B, 0, stream>>>((const float4*)embed_item,
                                                  (float4*)out, I * D / 4);

    // ---- 5 atomic-free pull SpMMs; item-direction passes fuse the combine ----
    auto pg = [&](int N) { return cdiv(N, B / 32); };
    // g1_users = spmm_ui(embed_item)
    pull_kernel_async<<<pg(U), B, 0, stream>>>(U, off_u, csr_u,
                                               embed_item, g1u, nullptr, 0.f);
    // g1_items = spmm_iu(embed_user);  out += 0.5 * g1_items
    pull_kernel_async<<<pg(I), B, 0, stream>>>(I, off_i, csr_i,
                                               embed_user, g1i, out, 0.5f);
    // g2_users = spmm_ui(g1_items)
    pull_kernel_async<<<pg(U), B, 0, stream>>>(U, off_u, csr_u,
                                               g1i, g2u, nullptr, 0.f);
    // g2_items = spmm_iu(g1_users);  out += (1/3) * g2_items  (never stored)
    pull_kernel_async<<<pg(I), B, 0, stream>>>(I, off_i, csr_i,
                                               g1u, nullptr, out, 1.0f / 3.0f);
    // g3_items = spmm_iu(g2_users);  out += 0.25 * g3_items  (never stored)
    pull_kernel_async<<<pg(I), B, 0, stream>>>(I, off_i, csr_i,
                                               g2u, nullptr, out, 0.25f);
}